// VectorQuantizer_58557584113930
// MI455X (gfx1250) — compile-verified
//
#include <hip/hip_runtime.h>
#include <hip/hip_bf16.h>
#include <stdint.h>

// ---------------------------------------------------------------------------
// VQ argmin:  dist(n,j) = ||z_n||^2 + ||e_j||^2 - 2 z_n.e_j
// ||z_n||^2 is constant per row -> argmin over  ||e_j||^2 - 2 * (z e^T)[n,j].
// The 8192x8192x256 f32 GEMM runs on V_WMMA_F32_16X16X4_F32 (keeps f32
// numerics for the argmin decision; bf16/f16 would perturb tiny e-values).
// ---------------------------------------------------------------------------

typedef float v2f __attribute__((ext_vector_type(2)));
typedef float v8f __attribute__((ext_vector_type(8)));

#define N_EMBED   8192
#define EMBED_DIM 256
#define ZQ_ELEMS  (8 * 256 * 32 * 32)   // 2,097,152 floats, indices follow
#define ZSTRIDE   260                    // 260 mod 64 = 4 -> bank-conflict-free A reads

// --- ||e_j||^2 : one wave per codebook row, coalesced lanes + shfl reduce ---
__global__ __launch_bounds__(256) void vq_enorm_kernel(const float* __restrict__ emb,
                                                       float* __restrict__ enorm) {
    const int gid  = blockIdx.x * blockDim.x + threadIdx.x;
    const int j    = gid >> 5;
    const int lane = threadIdx.x & 31;
    const float* row = emb + (size_t)j * EMBED_DIM;
    float s = 0.f;
#pragma unroll
    for (int c = 0; c < EMBED_DIM; c += 32) {
        float x = row[c + lane];
        s = __builtin_fmaf(x, x, s);
    }
#pragma unroll
    for (int m = 16; m; m >>= 1) s += __shfl_xor(s, m, 32);
    if (lane == 0) enorm[j] = s;
}

// --- main kernel: 256 WGs x 256 threads (8 waves); WG owns 32 z rows -------
__global__ __launch_bounds__(256) void vq_main_kernel(const float* __restrict__ hidden,
                                                      const float* __restrict__ emb,
                                                      const float* __restrict__ enorm,
                                                      float* __restrict__ out) {
    __shared__ float    zt[32 * ZSTRIDE];   // z tile, K-major per row, padded
    __shared__ float    wmin[8][32];        // per-wave row minima
    __shared__ unsigned widx[8][32];
    __shared__ unsigned fidx[32];

    const int tid    = threadIdx.x;
    const int wave   = tid >> 5;
    const int lane   = tid & 31;
    const int lane15 = lane & 15;
    const int half   = lane >> 4;           // selects K pair {0,1} vs {2,3}

    const int n0  = blockIdx.x * 32;        // 32 consecutive flattened pixels
    const int b   = n0 >> 10;               // 1024 pixels per batch image
    const int hw0 = n0 & 1023;
    const float* hbase = hidden + (size_t)b * (EMBED_DIM * 1024) + hw0;

    // Stage z tile: zt[r][c] = hidden[b, c, hw0+r].  32 consecutive threads
    // (same c, r=0..31) read 128 contiguous bytes -> fully coalesced.
    {
        const int r  = tid & 31;
        const int c0 = tid >> 5;
#pragma unroll
        for (int i = 0; i < 32; ++i) {
            const int c = c0 + 8 * i;
            zt[r * ZSTRIDE + c] = hbase[(size_t)c * 1024 + r];
        }
    }
    __syncthreads();

    // Running per-lane argmin. C/D layout: lane = N (mod 16), vgpr v = row M
    // (low half M=v, high half M=v+8); acc0 covers tile rows 0-15, acc1 16-31.
    float    minv0[8], minv1[8];
    unsigned mini0[8], mini1[8];
#pragma unroll
    for (int v = 0; v < 8; ++v) {
        minv0[v] = minv1[v] = __builtin_inff();
        mini0[v] = mini1[v] = 0u;
    }

    for (int jt = wave; jt < N_EMBED / 16; jt += 8) {
        const int j = jt * 16 + lane15;                       // this lane's N column
        const float* eb = emb + (size_t)j * EMBED_DIM + 2 * half;
        v8f acc0 = {0.f, 0.f, 0.f, 0.f, 0.f, 0.f, 0.f, 0.f};
        v8f acc1 = {0.f, 0.f, 0.f, 0.f, 0.f, 0.f, 0.f, 0.f};
#pragma unroll 8
        for (int k0 = 0; k0 < EMBED_DIM; k0 += 4) {
            // B fragment (K x N = 4x16): lane -> N, half -> K pair, one b64 load.
            const v2f bf = *(const v2f*)(eb + k0);
            // A fragments (M x K = 16x4) from padded LDS: conflict-free.
            const v2f a0 = *(const v2f*)(&zt[lane15 * ZSTRIDE + k0 + 2 * half]);
            const v2f a1 = *(const v2f*)(&zt[(16 + lane15) * ZSTRIDE + k0 + 2 * half]);
            acc0 = __builtin_amdgcn_wmma_f32_16x16x4_f32(false, a0, false, bf,
                                                         (short)0, acc0, false, false);
            acc1 = __builtin_amdgcn_wmma_f32_16x16x4_f32(false, a1, false, bf,
                                                         (short)0, acc1, false, false);
        }
        const float ej = enorm[j];
#pragma unroll
        for (int v = 0; v < 8; ++v) {
            const float d0 = __builtin_fmaf(-2.f, acc0[v], ej);
            const float d1 = __builtin_fmaf(-2.f, acc1[v], ej);
            const bool u0 = d0 < minv0[v];
            minv0[v] = u0 ? d0 : minv0[v];
            mini0[v] = u0 ? (unsigned)j : mini0[v];
            const bool u1 = d1 < minv1[v];
            minv1[v] = u1 ? d1 : minv1[v];
            mini1[v] = u1 ? (unsigned)j : mini1[v];
        }
    }

    // Reduce across the 16 N-lanes of each half (xor masks 1..8 stay in-half).
#pragma unroll
    for (int v = 0; v < 8; ++v) {
#pragma unroll
        for (int m = 1; m <= 8; m <<= 1) {
            float    ov = __shfl_xor(minv0[v], m, 32);
            unsigned oi = __shfl_xor(mini0[v], m, 32);
            if (ov < minv0[v] || (ov == minv0[v] && oi < mini0[v])) { minv0[v] = ov; mini0[v] = oi; }
            ov = __shfl_xor(minv1[v], m, 32);
            oi = __shfl_xor(mini1[v], m, 32);
            if (ov < minv1[v] || (ov == minv1[v] && oi < mini1[v])) { minv1[v] = ov; mini1[v] = oi; }
        }
        if (lane == 0) {                 // half 0 holds rows v and 16+v
            wmin[wave][v]      = minv0[v];  widx[wave][v]      = mini0[v];
            wmin[wave][16 + v] = minv1[v];  widx[wave][16 + v] = mini1[v];
        }
        if (lane == 16) {                // half 1 holds rows 8+v and 24+v
            wmin[wave][8 + v]  = minv0[v];  widx[wave][8 + v]  = mini0[v];
            wmin[wave][24 + v] = minv1[v];  widx[wave][24 + v] = mini1[v];
        }
    }
    __syncthreads();

    // Combine the 8 waves (each covered a disjoint j subset) per row.
    if (tid < 32) {
        float    bv = wmin[0][tid];
        unsigned bi = widx[0][tid];
#pragma unroll
        for (int w = 1; w < 8; ++w) {
            const float    ov = wmin[w][tid];
            const unsigned oi = widx[w][tid];
            if (ov < bv || (ov == bv && oi < bi)) { bv = ov; bi = oi; }
        }
        fidx[tid] = bi;
        out[ZQ_ELEMS + n0 + tid] = (float)bi;   // min_encoding_indices
    }
    __syncthreads();

    // z_q gather-write back into (B,C,H,W): coalesced 128B stores per c.
    {
        const int r  = tid & 31;
        const int c0 = tid >> 5;
        const unsigned idx = fidx[r];
        const float* erow = emb + (size_t)idx * EMBED_DIM;
        float* obase = out + (size_t)b * (EMBED_DIM * 1024) + hw0 + r;
#pragma unroll
        for (int i = 0; i < 32; ++i) {
            const int c = c0 + 8 * i;
            obase[(size_t)c * 1024] = erow[c];
        }
    }
}

extern "C" void kernel_launch(void* const* d_in, const int* in_sizes, int n_in,
                              void* d_out, int out_size, void* d_ws, size_t ws_size,
                              hipStream_t stream) {
    const float* hidden = (const float*)d_in[0];   // (8,256,32,32) f32
    const float* emb    = (const float*)d_in[1];   // (8192,256)    f32
    float* out   = (float*)d_out;                  // z_q (2,097,152) ++ indices (8192)
    float* enorm = (float*)d_ws;                   // 8192 floats of scratch

    // 8192 codebook rows, one wave each -> 1024 blocks of 8 waves.
    vq_enorm_kernel<<<N_EMBED / 8, 256, 0, stream>>>(emb, enorm);
    // 8192 pixels / 32 rows per WG -> 256 workgroups of 8 waves.
    vq_main_kernel<<<256, 256, 0, stream>>>(hidden, emb, enorm, out);
}